// MoEConv_65180423685461
// MI455X (gfx1250) — compile-verified
//
#include <hip/hip_runtime.h>
#include <hip/hip_bf16.h>

#define N_EXPERT 5
#define OUT_CH   128
#define IN_CH    128
#define HW       3136   // 56*56
#define IMG_W    56
#define KTAPS    9

#define CH_CHUNK 32                       // input channels staged per LDS pass
#define N_CHUNK  (IN_CH / CH_CHUNK)       // 4
#define LDS_ROWW 58                       // 56 + 2 halo cols
#define LDS_ROWS 4                        // rows h0-1 .. h0+2
#define LDS_ELEMS (16 * LDS_ROWS * LDS_ROWW * 2) // [icPair][row][col][par] = 7424 f32
#define NTILES   7                        // 2 rows * 56 px = 112 = 7 * 16, exact

#if __has_builtin(__builtin_amdgcn_global_load_async_to_lds_b32)
#define HAVE_ASYNC_LDS 1
#define ASYNC_WAIT() asm volatile("s_wait_asynccnt 0" ::: "memory")
typedef __attribute__((address_space(1))) int* gint_p;
typedef __attribute__((address_space(1))) void* gvoid_p;
typedef __attribute__((address_space(3))) int* lint_p;
typedef __attribute__((address_space(3))) void* lvoid_p;
#else
#define HAVE_ASYNC_LDS 0
#define ASYNC_WAIT()
#endif

typedef float v2f __attribute__((ext_vector_type(2)));
typedef float v8f __attribute__((ext_vector_type(8)));

// ---------------------------------------------------------------------------
// Kernel 0: per-batch argmax over 5 expert scores (first-max wins).
// ---------------------------------------------------------------------------
__global__ void moe_argmax_kernel(const float* __restrict__ scores,
                                  int* __restrict__ experts, int batch) {
    int b = blockIdx.x * blockDim.x + threadIdx.x;
    if (b >= batch) return;
    const float* s = scores + (size_t)b * N_EXPERT;
    int best = 0;
    float bv = s[0];
    #pragma unroll
    for (int e = 1; e < N_EXPERT; ++e) {
        float v = s[e];
        if (v > bv) { bv = v; best = e; }
    }
    experts[b] = best;
}

// ---------------------------------------------------------------------------
// Kernel 1: repack weights [E*OC][IC][3][3] -> Wp[e][tap][ic/2][oc][2]
// so the WMMA A-operand (16 oc x 4 ic, f32) per lane is one b64 load:
// lane L holds M = L%16, K = (L/16)*2 + {0,1}.
// ---------------------------------------------------------------------------
__global__ void moe_repack_kernel(const float* __restrict__ weight,
                                  float* __restrict__ Wp, int n) {
    int idx = blockIdx.x * blockDim.x + threadIdx.x;
    if (idx >= n) return;
    int i = idx;
    int j   = i & 1;      i >>= 1;
    int oc  = i & 127;    i >>= 7;
    int icp = i & 63;     i >>= 6;
    int t   = i % KTAPS;  i /= KTAPS;
    int e   = i;
    int ic  = icp * 2 + j;
    size_t src = ((size_t)(e * OUT_CH + oc) * IN_CH + ic) * KTAPS + t;
    Wp[idx] = weight[src];
}

// ---------------------------------------------------------------------------
// Kernel 2: implicit-GEMM 3x3 conv (pad 1) via V_WMMA_F32_16X16X4_F32.
// Block = 256 threads (8 wave32) = 128 oc x two output rows (112 px = 7x16,
// zero wasted WMMA columns). Input double-buffered in LDS via
// GLOBAL_LOAD_ASYNC_TO_LDS_B32 so chunk k+1 staging overlaps chunk k compute;
// B fed by ds_load_b64; A amortized over 7 WMMAs per load.
// ---------------------------------------------------------------------------
__global__ void __launch_bounds__(256)
moe_conv_wmma_kernel(const float* __restrict__ x,
                     const float* __restrict__ Wp,
                     const int* __restrict__ experts,
                     float* __restrict__ out) {
    __shared__ float lds[2 * LDS_ELEMS];

    const int b      = blockIdx.z;
    const int h0     = blockIdx.x * 2;               // output row pair base
    const int wave   = threadIdx.x >> 5;
    const int lane   = threadIdx.x & 31;
    const int laneM  = lane & 15;                    // M (A) / N (B) index
    const int laneKh = lane >> 4;                    // K half
    const int ocBase = wave * 16;

    const int e = experts[b];
    const float* xb = x + (size_t)b * IN_CH * HW;

    // Per-N-tile (row, col) of this lane's output pixel; chunk-invariant
    // LDS offset contribution: (row*58 + col)*2.
    int off[NTILES];
    #pragma unroll
    for (int nt = 0; nt < NTILES; ++nt) {
        int p  = nt * 16 + laneM;        // 0..111
        int ro = p / IMG_W;              // 0 or 1
        int co = p - ro * IMG_W;
        off[nt] = (ro * LDS_ROWW + co) * 2;
    }

    // ---- Pre-zero halo cells of BOTH buffers (positions are chunk-invariant;
    //      async copies only ever write interior cells, so once is enough).
    for (int i = threadIdx.x; i < LDS_ELEMS; i += 256) {
        int q2 = i >> 1;
        int c  = q2 % LDS_ROWW;  q2 /= LDS_ROWW;
        int r  = q2 & 3;
        int y  = h0 - 1 + r;
        int xx = c - 1;
        if (!((unsigned)y < (unsigned)IMG_W && (unsigned)xx < (unsigned)IMG_W)) {
            lds[i] = 0.0f;
            lds[LDS_ELEMS + i] = 0.0f;
        }
    }

    // ---- Stage-issue: copy interior of [32 ic][4 rows][56 cols] for `chunk`
    //      into buffer `buf`, interleaved [icPair][row][col][parity].
    auto stage = [&](int buf, int chunk) {
        const int c0 = chunk * CH_CHUNK;
        float* dst = &lds[buf * LDS_ELEMS];
        for (int i = threadIdx.x; i < LDS_ELEMS; i += 256) {
            int q2  = i;
            int par = q2 & 1;           q2 >>= 1;
            int c   = q2 % LDS_ROWW;    q2 /= LDS_ROWW;
            int r   = q2 & 3;           q2 >>= 2;
            int icp = q2;               // 0..15
            int y   = h0 - 1 + r;
            int xx  = c - 1;
            if ((unsigned)y < (unsigned)IMG_W && (unsigned)xx < (unsigned)IMG_W) {
                const float* g = xb + (size_t)(c0 + icp * 2 + par) * HW +
                                 y * IMG_W + xx;
#if HAVE_ASYNC_LDS
                __builtin_amdgcn_global_load_async_to_lds_b32(
                    (gint_p)(gvoid_p)(uintptr_t)g,
                    (lint_p)(lvoid_p)&dst[i],
                    /*offset=*/0, /*cpol=*/0);
#else
                dst[i] = *g;
#endif
            }
        }
    };

    v8f acc[NTILES];
    #pragma unroll
    for (int nt = 0; nt < NTILES; ++nt) acc[nt] = (v8f)0.0f;

    stage(0, 0);

    for (int chunk = 0; chunk < N_CHUNK; ++chunk) {
        ASYNC_WAIT();          // own wave's async copies for this chunk done
        __syncthreads();       // all waves staged; previous compute finished

        if (chunk + 1 < N_CHUNK)
            stage((chunk + 1) & 1, chunk + 1);   // overlaps compute below

        const float* buf = &lds[(chunk & 1) * LDS_ELEMS];

        // ---- Compute: 9 taps x 8 ic-quads over this chunk.
        #pragma unroll
        for (int t = 0; t < KTAPS; ++t) {
            const int rr = t / 3;       // tap row (dh+1)
            const int cc = t % 3;       // tap col (dw+1)
            const float* wA = Wp +
                ((size_t)((e * KTAPS + t) * 64 + chunk * 16 + laneKh) * OUT_CH +
                 (ocBase + laneM)) * 2;

            #pragma unroll 2
            for (int q = 0; q < 8; ++q) {
                v2f a = *(const v2f*)(wA + (size_t)q * (2 * OUT_CH * 2));
                const int bBase =
                    (((q * 2 + laneKh) * LDS_ROWS + rr) * LDS_ROWW + cc) * 2;
                #pragma unroll
                for (int nt = 0; nt < NTILES; ++nt) {
                    v2f bb = *(const v2f*)&buf[bBase + off[nt]];
                    acc[nt] = __builtin_amdgcn_wmma_f32_16x16x4_f32(
                        /*neg_a=*/false, a, /*neg_b=*/false, bb,
                        /*c_mod=*/(short)0, acc[nt],
                        /*reuse_a=*/false, /*reuse_b=*/false);
                }
            }
        }
    }

    // ---- Store: D layout -> VGPR r: lanes 0-15 M=r, lanes 16-31 M=r+8.
    float* ob = out + (size_t)b * OUT_CH * HW + (size_t)h0 * IMG_W;
    #pragma unroll
    for (int nt = 0; nt < NTILES; ++nt) {
        int p = nt * 16 + laneM;         // 0..111 within the row pair, all valid
        #pragma unroll
        for (int r = 0; r < 8; ++r) {
            int oc = ocBase + laneKh * 8 + r;
            ob[(size_t)oc * HW + p] = acc[nt][r];
        }
    }
}

extern "C" void kernel_launch(void* const* d_in, const int* in_sizes, int n_in,
                              void* d_out, int out_size, void* d_ws, size_t ws_size,
                              hipStream_t stream) {
    const float* x      = (const float*)d_in[0];   // [32,128,56,56]
    const float* scores = (const float*)d_in[1];   // [32,5]
    const float* weight = (const float*)d_in[2];   // [640,128,3,3]
    float*       out    = (float*)d_out;           // [32,128,56,56]

    const int batch = in_sizes[1] / N_EXPERT;      // 32

    int*   experts = (int*)d_ws;
    float* Wp      = (float*)((char*)d_ws + 256);  // 5*9*64*128*2 f32 ≈ 2.95 MB

    moe_argmax_kernel<<<1, 32, 0, stream>>>(scores, experts, batch);

    const int nrep = N_EXPERT * KTAPS * 64 * OUT_CH * 2;
    moe_repack_kernel<<<(nrep + 255) / 256, 256, 0, stream>>>(weight, Wp, nrep);

    dim3 grid(IMG_W / 2, 1, batch);   // 28 x 1 x 32
    moe_conv_wmma_kernel<<<grid, 256, 0, stream>>>(x, Wp, experts, out);
}